// GraphConvLayer_69458211111561
// MI455X (gfx1250) — compile-verified
//
#include <hip/hip_runtime.h>
#include <hip/hip_bf16.h>

typedef __attribute__((ext_vector_type(2))) float v2f;
typedef __attribute__((ext_vector_type(8))) float v8f;

// ---------------------------------------------------------------------------
// Stage 2: degree counts. One thread per edge, two atomic f32 adds.
// ---------------------------------------------------------------------------
__global__ void gc_degree_kernel(const int* __restrict__ src,
                                 const int* __restrict__ dst,
                                 float* __restrict__ outdeg,
                                 float* __restrict__ indeg,
                                 int E) {
    int e = blockIdx.x * blockDim.x + threadIdx.x;
    if (e >= E) return;
    atomicAdd(&outdeg[src[e]], 1.0f);
    atomicAdd(&indeg[dst[e]], 1.0f);
}

// ---------------------------------------------------------------------------
// Stage 3: gather + normalize + scatter-sum. One wave32 per edge; each lane
// handles 2 consecutive floats of the 64-dim feature row.
// ---------------------------------------------------------------------------
__global__ void gc_scatter_kernel(const float* __restrict__ feats,
                                  const int* __restrict__ src,
                                  const int* __restrict__ dst,
                                  const float* __restrict__ outdeg,
                                  float* __restrict__ agg,
                                  int E) {
    int tid  = blockIdx.x * blockDim.x + threadIdx.x;
    int e    = tid >> 5;
    int lane = tid & 31;
    if (e >= E) return;

    int s = src[e];
    int d = dst[e];
    float deg = outdeg[s];                       // broadcast load (L0/L2 hit)
    float ns  = deg > 0.0f ? rsqrtf(deg) : 0.0f;

    const v2f* frow = reinterpret_cast<const v2f*>(feats + (size_t)s * 64) + lane;
    v2f v = *frow;

    float* arow = agg + (size_t)d * 64 + lane * 2;
    atomicAdd(arow,     v.x * ns);
    atomicAdd(arow + 1, v.y * ns);
}

// ---------------------------------------------------------------------------
// Stage 4: out = relu((agg * norm_dst) @ W + b) with V_WMMA_F32_16X16X4_F32.
// One wave computes one 16x64 output tile (4 N-tiles of 16x16, 16 K-steps).
//
// 32-bit A 16x4 layout: lanes 0-15 hold M=0..15 with K={0,1} in VGPR{0,1};
//                       lanes 16-31 hold M=0..15 with K={2,3}.
// 32-bit B 4x16 layout: mirrored (lane = column N, VGPRs = K values).
// 32-bit C/D 16x16:     VGPR r -> row (half*8 + r), lane%16 -> column.
// ---------------------------------------------------------------------------
__global__ void gc_gemm_wmma_kernel(const float* __restrict__ agg,
                                    const float* __restrict__ indeg,
                                    const float* __restrict__ W,
                                    const float* __restrict__ bias,
                                    float* __restrict__ out,
                                    int n_nodes) {
    int wave = (blockIdx.x * blockDim.x + threadIdx.x) >> 5;
    int lane = threadIdx.x & 31;
    int m0 = wave * 16;
    if (m0 >= n_nodes) return;                   // wave-uniform: EXEC stays all-1s

    int half = lane >> 4;                        // 0: K={0,1}, 1: K={2,3}
    int ml   = lane & 15;                        // A row / B,C,D column within tile
    int m    = m0 + ml;

    float deg = indeg[m];
    float nd  = deg > 0.0f ? rsqrtf(deg) : 0.0f;

    // Load the entire 16x64 A tile (this wave's rows), scaled, into registers.
    v2f A[16];
    const float* arow = agg + (size_t)m * 64;
    #pragma unroll
    for (int kb = 0; kb < 16; ++kb) {
        int k = kb * 4 + half * 2;
        v2f a;
        a.x = arow[k]     * nd;
        a.y = arow[k + 1] * nd;
        A[kb] = a;
    }

    // 4 column tiles of the 64-wide output.
    #pragma unroll
    for (int nt = 0; nt < 4; ++nt) {
        int ncol = nt * 16 + ml;
        v8f acc = {};
        #pragma unroll
        for (int kb = 0; kb < 16; ++kb) {
            int k = kb * 4 + half * 2;
            v2f b;
            b.x = W[(size_t)k       * 64 + ncol];
            b.y = W[(size_t)(k + 1) * 64 + ncol];
            acc = __builtin_amdgcn_wmma_f32_16x16x4_f32(
                /*neg_a=*/false, A[kb],
                /*neg_b=*/false, b,
                /*c_mod=*/(short)0, acc,
                /*reuse_a=*/false, /*reuse_b=*/false);
        }
        float bv = bias[ncol];
        #pragma unroll
        for (int r = 0; r < 8; ++r) {
            int row = m0 + half * 8 + r;
            float v = acc[r] + bv;
            out[(size_t)row * 64 + ncol] = v > 0.0f ? v : 0.0f;
        }
    }
}

// ---------------------------------------------------------------------------
// Launcher. Inputs: feats[N*64] f32, src[E] i32, dst[E] i32, W[64*64] f32,
// b[64] f32. Output: [N*64] f32. Workspace: agg[N*64] + outdeg[N] + indeg[N].
// ---------------------------------------------------------------------------
extern "C" void kernel_launch(void* const* d_in, const int* in_sizes, int n_in,
                              void* d_out, int out_size, void* d_ws, size_t ws_size,
                              hipStream_t stream) {
    const float* feats = (const float*)d_in[0];
    const int*   src   = (const int*)d_in[1];
    const int*   dst   = (const int*)d_in[2];
    const float* W     = (const float*)d_in[3];
    const float* bias  = (const float*)d_in[4];

    int E = in_sizes[1];
    int N = in_sizes[0] / 64;

    float* agg    = (float*)d_ws;
    float* outdeg = agg + (size_t)N * 64;
    float* indeg  = outdeg + N;

    size_t zero_bytes = ((size_t)N * 64 + 2 * (size_t)N) * sizeof(float);
    hipMemsetAsync(d_ws, 0, zero_bytes, stream);

    int tb = 256;
    int deg_blocks = (E + tb - 1) / tb;
    gc_degree_kernel<<<deg_blocks, tb, 0, stream>>>(src, dst, outdeg, indeg, E);

    long long sc_threads = (long long)E * 32;
    int sc_blocks = (int)((sc_threads + tb - 1) / tb);
    gc_scatter_kernel<<<sc_blocks, tb, 0, stream>>>(feats, src, dst, outdeg, agg, E);

    int tiles = (N + 15) / 16;                 // one wave per 16-row tile
    int gemm_blocks = (tiles * 32 + tb - 1) / tb;
    gc_gemm_wmma_kernel<<<gemm_blocks, tb, 0, stream>>>(agg, indeg, W, bias,
                                                        (float*)d_out, N);
}